// GCN_36653250904445
// MI455X (gfx1250) — compile-verified
//
#include <hip/hip_runtime.h>
#include <math.h>

// ---------------------------------------------------------------------------
// 3-layer GCN on MI455X (gfx1250).
//  - GEMMs via V_WMMA_F32_16X16X4_F32 (fp32-exact; compute is not the bottleneck)
//  - Edge aggregation via global_atomic_add_f32 (L2-resident: h/agg = 25.6MB << 192MB L2)
//  - wave32-aware log_softmax reductions (__shfl_xor width 32)
// ---------------------------------------------------------------------------

typedef __attribute__((ext_vector_type(2))) float v2f;
typedef __attribute__((ext_vector_type(8))) float v8f;

#define THREADS 256

// ---------------- degree / norm ----------------
__global__ void k_degree(const int* __restrict__ dst, unsigned* __restrict__ cnt, int E) {
    int e = blockIdx.x * blockDim.x + threadIdx.x;
    if (e < E) atomicAdd(&cnt[dst[e]], 1u);
}

__global__ void k_dinv(const unsigned* __restrict__ cnt, float* __restrict__ dinv, int n) {
    int i = blockIdx.x * blockDim.x + threadIdx.x;
    if (i < n) dinv[i] = rsqrtf((float)cnt[i] + 1.0f);
}

// ---------------- WMMA GEMM: H[n,OUT] = A[n,K] @ W[K,OUT] ----------------
// One wave per 16x16 output tile. K stepped by 4 via V_WMMA_F32_16X16X4_F32.
// A fragment (16x4 f32): lanes 0-15 rows M=0..15 @ K={k0,k0+1}; lanes 16-31 @ K={k0+2,k0+3}.
// B fragment (4x16 f32): mirrored (K rows striped across lane halves, N across lanes).
// C/D (16x16 f32, 8 VGPRs): lane<16 -> M=v, N=lane ; lane>=16 -> M=v+8, N=lane-16.
//
// Out-of-range rows/cols are CLAMPED, not zeroed: a garbage A-row m only affects
// D-row m and a garbage B-col n only affects D-col n, neither of which is stored.
// This keeps all fragment loads unconditional (plain global_load_b64 / ds_load_2addr),
// avoids per-element EXEC masking, and keeps EXEC all-1s for WMMA.
template <int K, int OUT>
__global__ __launch_bounds__(THREADS)
void k_gemm_wmma(const float* __restrict__ A, const float* __restrict__ W,
                 float* __restrict__ Hout, int nrows) {
    __shared__ float sW[K * OUT];
    for (int i = threadIdx.x; i < K * OUT; i += THREADS) sW[i] = W[i];
    __syncthreads();

    constexpr int CT = (OUT + 15) / 16;
    const int rowTiles = (nrows + 15) >> 4;
    const int tile = blockIdx.x * (THREADS / 32) + (threadIdx.x >> 5);
    if (tile >= rowTiles * CT) return;

    const int rowTile = tile / CT;
    const int colTile = tile % CT;
    const int lane = threadIdx.x & 31;
    const int half = lane >> 4;      // selects K-pair (and M-half on store)
    const int mn   = lane & 15;

    const int row0 = rowTile * 16;
    const int col0 = colTile * 16;
    int r = row0 + mn; if (r >= nrows) r = nrows - 1;  // clamp (never stored)
    int c = col0 + mn; if (c >= OUT)  c = OUT - 1;     // clamp (never stored)

    const float* ap = A + (size_t)r * K + 2 * half;    // this lane's K-pair base
    const float* bp = sW + c;

    v8f acc = {};
#pragma unroll
    for (int k0 = 0; k0 < K; k0 += 4) {
        const v2f a = *(const v2f*)(ap + k0);          // unconditional b64 load
        v2f b;
        b.x = bp[(k0 + 2 * half) * OUT];               // ds_load_2addr pair
        b.y = bp[(k0 + 2 * half + 1) * OUT];
        acc = __builtin_amdgcn_wmma_f32_16x16x4_f32(
            /*neg_a=*/false, a, /*neg_b=*/false, b,
            /*c_mod=*/(short)0, acc, /*reuse_a=*/false, /*reuse_b=*/false);
    }

    const int cs = col0 + mn;                          // unclamped for store guard
#pragma unroll
    for (int v = 0; v < 8; ++v) {
        const int rr = row0 + v + 8 * half;
        if (rr < nrows && cs < OUT) Hout[(size_t)rr * OUT + cs] = acc[v];
    }
}

// ---------------- edge scatter: agg[dst] += h[src] * dinv[src]*dinv[dst] ----------------
template <int OUT>
__global__ void k_scatter(const int* __restrict__ src, const int* __restrict__ dst,
                          const float* __restrict__ dinv, const float* __restrict__ h,
                          float* __restrict__ agg, int E) {
    constexpr int TPE = OUT / 4;  // threads per edge (float4 per thread)
    const long long g = (long long)blockIdx.x * blockDim.x + threadIdx.x;
    const long long e = g / TPE;
    if (e >= E) return;
    const int fo = (int)(g % TPE) * 4;
    const int s = src[e];
    const int d = dst[e];
    const float nrm = dinv[s] * dinv[d];
    const float4 m = *(const float4*)(h + (size_t)s * OUT + fo);
    float* ap = agg + (size_t)d * OUT + fo;
    atomicAdd(ap + 0, m.x * nrm);
    atomicAdd(ap + 1, m.y * nrm);
    atomicAdd(ap + 2, m.z * nrm);
    atomicAdd(ap + 3, m.w * nrm);
}

// ---------------- epilogue: relu(agg + h*dinv^2 + b) ----------------
__global__ void k_finish_relu(const float* __restrict__ agg, const float* __restrict__ h,
                              const float* __restrict__ dinv, const float* __restrict__ b,
                              float* __restrict__ out, int n, int F) {
    const long long g = (long long)blockIdx.x * blockDim.x + threadIdx.x;
    if (g >= (long long)n * F) return;
    const int node = (int)(g / F);
    const int f = (int)(g % F);
    const float di = dinv[node];
    const float v = agg[g] + h[g] * di * di + b[f];
    out[g] = v > 0.0f ? v : 0.0f;
}

// ---------------- epilogue: log_softmax(agg + h*dinv^2 + b), one wave32 per row ----------------
template <int OUT>
__global__ __launch_bounds__(256)
void k_finish_lsm(const float* __restrict__ agg, const float* __restrict__ h,
                  const float* __restrict__ dinv, const float* __restrict__ b,
                  float* __restrict__ out, int n) {
    const int wave = threadIdx.x >> 5;
    const int lane = threadIdx.x & 31;
    const int node = blockIdx.x * 8 + wave;
    if (node >= n) return;
    const float di = dinv[node];
    const float di2 = di * di;
    const float* ag = agg + (size_t)node * OUT;
    const float* hh = h + (size_t)node * OUT;
    const int f0 = lane, f1 = lane + 32;
    float v0 = -INFINITY, v1 = -INFINITY;
    if (f0 < OUT) v0 = ag[f0] + hh[f0] * di2 + b[f0];
    if (f1 < OUT) v1 = ag[f1] + hh[f1] * di2 + b[f1];
    float mx = fmaxf(v0, v1);
    for (int m = 16; m >= 1; m >>= 1) mx = fmaxf(mx, __shfl_xor(mx, m, 32));
    float s = 0.0f;
    if (f0 < OUT) s += expf(v0 - mx);
    if (f1 < OUT) s += expf(v1 - mx);
    for (int m = 16; m >= 1; m >>= 1) s += __shfl_xor(s, m, 32);
    const float lse = logf(s) + mx;
    if (f0 < OUT) out[(size_t)node * OUT + f0] = v0 - lse;
    if (f1 < OUT) out[(size_t)node * OUT + f1] = v1 - lse;
}

// ---------------------------------------------------------------------------
extern "C" void kernel_launch(void* const* d_in, const int* in_sizes, int n_in,
                              void* d_out, int out_size, void* d_ws, size_t ws_size,
                              hipStream_t stream) {
    constexpr int F_IN = 128, H = 64, C = 40;
    const float* x  = (const float*)d_in[0];
    const int*   ei = (const int*)d_in[1];   // edge_index [2, E] (src row, dst row)
    const float* W1 = (const float*)d_in[2];
    const float* b1 = (const float*)d_in[3];
    const float* W2 = (const float*)d_in[4];
    const float* b2 = (const float*)d_in[5];
    const float* W3 = (const float*)d_in[6];
    const float* b3 = (const float*)d_in[7];
    float* out = (float*)d_out;

    const int N = in_sizes[0] / F_IN;   // 100000
    const int E = in_sizes[1] / 2;      // 1600000
    const int* srcA = ei;
    const int* dstA = ei + E;

    // workspace layout (256B-aligned slices)
    char* ws = (char*)d_ws;
    size_t off = 0;
    auto alloc = [&](size_t bytes) -> void* {
        void* p = ws + off;
        off += (bytes + 255) & ~(size_t)255;
        return p;
    };
    unsigned* cnt  = (unsigned*)alloc((size_t)N * sizeof(unsigned));
    float*    dinv = (float*)alloc((size_t)N * sizeof(float));
    float*    bufH = (float*)alloc((size_t)N * H * sizeof(float)); // h = act @ W
    float*    bufA = (float*)alloc((size_t)N * H * sizeof(float)); // agg
    float*    bufC = (float*)alloc((size_t)N * H * sizeof(float)); // activation
    (void)ws_size; (void)n_in; (void)out_size;

    // symmetric-norm degrees
    hipMemsetAsync(cnt, 0, (size_t)N * sizeof(unsigned), stream);
    k_degree<<<(E + 255) / 256, 256, 0, stream>>>(dstA, cnt, E);
    k_dinv<<<(N + 255) / 256, 256, 0, stream>>>(cnt, dinv, N);

    const int rowTiles = (N + 15) / 16;

    // -------- layer 1: 128 -> 64, relu --------
    {
        const int tiles = rowTiles * (H / 16);
        k_gemm_wmma<F_IN, H><<<(tiles + 7) / 8, THREADS, 0, stream>>>(x, W1, bufH, N);
        hipMemsetAsync(bufA, 0, (size_t)N * H * sizeof(float), stream);
        const long long tot = (long long)E * (H / 4);
        k_scatter<H><<<(unsigned)((tot + 255) / 256), 256, 0, stream>>>(srcA, dstA, dinv, bufH, bufA, E);
        const long long nf = (long long)N * H;
        k_finish_relu<<<(unsigned)((nf + 255) / 256), 256, 0, stream>>>(bufA, bufH, dinv, b1, bufC, N, H);
    }
    // -------- layer 2: 64 -> 64, log_softmax --------
    {
        const int tiles = rowTiles * (H / 16);
        k_gemm_wmma<H, H><<<(tiles + 7) / 8, THREADS, 0, stream>>>(bufC, W2, bufH, N);
        hipMemsetAsync(bufA, 0, (size_t)N * H * sizeof(float), stream);
        const long long tot = (long long)E * (H / 4);
        k_scatter<H><<<(unsigned)((tot + 255) / 256), 256, 0, stream>>>(srcA, dstA, dinv, bufH, bufA, E);
        k_finish_lsm<H><<<(N + 7) / 8, 256, 0, stream>>>(bufA, bufH, dinv, b2, bufC, N);
    }
    // -------- layer 3: 64 -> 40, log_softmax --------
    {
        const int tiles = rowTiles * ((C + 15) / 16);
        k_gemm_wmma<H, C><<<(tiles + 7) / 8, THREADS, 0, stream>>>(bufC, W3, bufH, N);
        hipMemsetAsync(bufA, 0, (size_t)N * C * sizeof(float), stream);
        const long long tot = (long long)E * (C / 4);
        k_scatter<C><<<(unsigned)((tot + 255) / 256), 256, 0, stream>>>(srcA, dstA, dinv, bufH, bufA, E);
        k_finish_lsm<C><<<(N + 7) / 8, 256, 0, stream>>>(bufA, bufH, dinv, b3, out, N);
    }
}